// MEO_88055419502758
// MI455X (gfx1250) — compile-verified
//
#include <hip/hip_runtime.h>

typedef float v2f __attribute__((ext_vector_type(2)));
typedef float v8f __attribute__((ext_vector_type(8)));
typedef int   v4i __attribute__((ext_vector_type(4)));

#define AS1 __attribute__((address_space(1)))
#define AS3 __attribute__((address_space(3)))

#if __has_builtin(__builtin_amdgcn_global_load_async_to_lds_b128) && \
    __has_builtin(__builtin_amdgcn_s_wait_asynccnt)
#define USE_ASYNC 1
#else
#define USE_ASYNC 0
#endif

#define E_   8
#define K_   4
#define IN_  768
#define OUT_ 768
#define B_   16
#define S_   512
#define WSZ  (OUT_ * IN_)          // 589824 per expert
#define YSZ  ((size_t)B_ * S_ * OUT_)

// ---------------------------------------------------------------------------
// Kernel 1: xm[b][j] = mean over s of x[b][s][j]
// ---------------------------------------------------------------------------
__global__ __launch_bounds__(IN_) void xm_kernel(const float* __restrict__ x,
                                                 float* __restrict__ xm) {
  const int b = blockIdx.x;
  const int j = threadIdx.x;
  const float* xb = x + (size_t)b * S_ * IN_ + j;
  float acc = 0.f;
  for (int s = 0; s < S_; ++s) acc += xb[(size_t)s * IN_];
  xm[b * IN_ + j] = acc * (1.0f / (float)S_);
}

// ---------------------------------------------------------------------------
// Kernel 2: logits -> top-4 softmax gates + cv^2 load-balancing loss
// ---------------------------------------------------------------------------
__global__ __launch_bounds__(128) void gate_kernel(const float* __restrict__ xm,
                                                   const float* __restrict__ wg,
                                                   float* __restrict__ gates_out,
                                                   float* __restrict__ loss_out) {
  __shared__ float s_logits[B_ * E_];
  __shared__ float s_gates[B_ * E_];
  const int t = threadIdx.x;
  {
    const int b = t >> 3, e = t & 7;
    float acc = 0.f;
    for (int j = 0; j < IN_; ++j) acc += xm[b * IN_ + j] * wg[j * E_ + e];
    s_logits[t] = acc;
    s_gates[t] = 0.f;
  }
  __syncthreads();
  if (t < B_) {
    const int b = t;
    float l[E_];
    for (int e = 0; e < E_; ++e) l[e] = s_logits[b * E_ + e];
    int   idx[K_];
    float vals[K_];
    bool  used[E_] = {};
    for (int r = 0; r < K_; ++r) {          // stable top-4 (first-index wins)
      float best = -3.4e38f; int bi = 0;
      for (int e = 0; e < E_; ++e)
        if (!used[e] && l[e] > best) { best = l[e]; bi = e; }
      used[bi] = true; idx[r] = bi; vals[r] = best;
    }
    const float mx = vals[0];
    float ex[K_], s = 0.f;
    for (int r = 0; r < K_; ++r) { ex[r] = expf(vals[r] - mx); s += ex[r]; }
    for (int r = 0; r < K_; ++r) s_gates[b * E_ + idx[r]] = ex[r] / s;
  }
  __syncthreads();
  if (t < B_ * E_) gates_out[t] = s_gates[t];
  if (t == 0) {
    float imp[E_], ld[E_];
    for (int e = 0; e < E_; ++e) { imp[e] = 0.f; ld[e] = 0.f; }
    for (int b = 0; b < B_; ++b)
      for (int e = 0; e < E_; ++e) {
        const float g = s_gates[b * E_ + e];
        imp[e] += g;
        ld[e]  += (g > 0.f) ? 1.f : 0.f;
      }
    float cv[2];
    for (int w = 0; w < 2; ++w) {
      const float* v = w ? ld : imp;
      float mean = 0.f;
      for (int e = 0; e < E_; ++e) mean += v[e];
      mean *= (1.0f / E_);
      float var = 0.f;
      for (int e = 0; e < E_; ++e) { const float d = v[e] - mean; var += d * d; }
      var *= (1.0f / (E_ - 1));
      cv[w] = var / (mean * mean + 1e-10f);
    }
    *loss_out = 0.01f * (cv[0] + cv[1]);
  }
}

// ---------------------------------------------------------------------------
// Kernel 3: generic mode-n product over a strided axis.
// out[e,a,i,c] = sum_t coef[e,i,t] * in[e,a,t,c]; optional fused subtraction.
// ---------------------------------------------------------------------------
__global__ __launch_bounds__(256) void mode_strided_kernel(
    const float* __restrict__ in, const float* __restrict__ sub,
    const float* __restrict__ coef, float* __restrict__ out,
    int A, int CN, int Csz) {
  __shared__ float s_coef[32 * 32];
  __shared__ float s_tile[32 * 32];
  const int e  = blockIdx.z;
  const int a  = blockIdx.y;
  const int c0 = blockIdx.x * 32;
  const int tid = threadIdx.x;

  for (int i = tid; i < CN * CN; i += 256) s_coef[i] = coef[e * CN * CN + i];

  const size_t ebase = (size_t)e * WSZ;
  for (int i = tid; i < CN * 32; i += 256) {
    const int tt = i >> 5, c = i & 31;
    const size_t li = ((size_t)a * CN + tt) * Csz + c0 + c;  // expert-local flat
    float v = in[ebase + li];
    if (sub) v -= sub[li];
    s_tile[tt * 32 + c] = v;
  }
  __syncthreads();

  const int c = tid & 31;
  for (int i = tid >> 5; i < CN; i += 8) {
    float acc = 0.f;
    for (int tt = 0; tt < CN; ++tt)
      acc += s_coef[i * CN + tt] * s_tile[tt * 32 + c];   // coef: wave broadcast
    out[ebase + ((size_t)a * CN + i) * Csz + c0 + c] = acc;
  }
}

// ---------------------------------------------------------------------------
// Kernel 4: mode product over the innermost contiguous axis (CN = 32).
// ---------------------------------------------------------------------------
__global__ __launch_bounds__(256) void mode_inner_kernel(
    const float* __restrict__ in, const float* __restrict__ coef,
    float* __restrict__ out) {
  __shared__ float s_coef[32 * 33];    // pad 33: conflict-free i-strided reads
  __shared__ float s_in[8 * 32];
  const int e  = blockIdx.y;
  const int a0 = blockIdx.x * 8;
  const int tid = threadIdx.x;

  for (int i = tid; i < 1024; i += 256)
    s_coef[(i >> 5) * 33 + (i & 31)] = coef[e * 1024 + i];
  {
    const int r = tid >> 5, m = tid & 31;
    s_in[r * 32 + m] = in[(size_t)e * WSZ + (size_t)(a0 + r) * 32 + m];
  }
  __syncthreads();

  const int r = tid >> 5, i = tid & 31;
  float acc = 0.f;
  for (int m = 0; m < 32; ++m) acc += s_coef[i * 33 + m] * s_in[r * 32 + m];
  out[(size_t)e * WSZ + (size_t)(a0 + r) * 32 + i] = acc;
}

// ---------------------------------------------------------------------------
// Kernel 5: ew[b] = res + 0.9 * sum_e gates[b,e] * rtw[e]
// ---------------------------------------------------------------------------
__global__ __launch_bounds__(256) void combine_kernel(
    const float* __restrict__ rtw, const float* __restrict__ res,
    const float* __restrict__ gates, float* __restrict__ ew) {
  const int b = blockIdx.y;
  const int idx = blockIdx.x * 256 + threadIdx.x;
  float sum = 0.f;
  for (int e = 0; e < E_; ++e)
    sum += gates[b * E_ + e] * rtw[(size_t)e * WSZ + idx];
  ew[(size_t)b * WSZ + idx] = res[idx] + 0.9f * sum;
}

// ---------------------------------------------------------------------------
// Kernel 6: main GEMM with FP32 WMMA, double-buffered LDS.
// y[b] (512x768) = x[b] (512x768) @ ew[b]^T; ew stored [N][K] (K contiguous).
// Block: 256 thr = 8 waves (4M x 2N); block tile 128x64, K-stage 64.
// Staging: async global->LDS copy (ASYNCcnt) when available, else
// register-prefetch software pipeline. One barrier per K-iteration.
// ---------------------------------------------------------------------------
#define LDP 68
__global__ __launch_bounds__(256) void gemm_kernel(const float* __restrict__ x,
                                                   const float* __restrict__ ew,
                                                   float* __restrict__ y) {
  __shared__ float lds_a[2][128 * LDP];
  __shared__ float lds_b[2][64 * LDP];
  const int b  = blockIdx.z;
  const int m0 = blockIdx.y * 128;
  const int n0 = blockIdx.x * 64;
  const float* Ag = x  + (size_t)b * S_ * IN_ + (size_t)m0 * IN_;
  const float* Bg = ew + (size_t)b * WSZ     + (size_t)n0 * IN_;

  const int tid   = threadIdx.x;
  const int lane  = tid & 31;
  const int wave  = tid >> 5;
  const int wm    = wave & 3;     // 0..3 -> 32-row band
  const int wn    = wave >> 2;    // 0..1 -> 32-col band
  const int mrow  = lane & 15;
  const int khalf = lane >> 4;    // lanes 0-15: K{0,1}; lanes 16-31: K{2,3}
  const int koff  = khalf * 2;

  const int a_row0 = (wm * 32 + mrow) * LDP;
  const int a_row1 = (wm * 32 + 16 + mrow) * LDP;
  const int b_row0 = (wn * 32 + mrow) * LDP;
  const int b_row1 = (wn * 32 + 16 + mrow) * LDP;

  v8f acc00 = {}, acc01 = {}, acc10 = {}, acc11 = {};

#if USE_ASYNC
  auto stage_async = [&](int k0, int buf) {
    #pragma unroll
    for (int i = 0; i < 8; ++i) {               // A: 128 rows x 64 K
      const int idx = tid + i * 256;
      const int r = idx >> 4;
      const int c = (idx & 15) << 2;
      __builtin_amdgcn_global_load_async_to_lds_b128(
          (AS1 v4i*)(uintptr_t)(Ag + (size_t)r * IN_ + k0 + c),
          (AS3 v4i*)(&lds_a[buf][r * LDP + c]), 0, 0);
    }
    #pragma unroll
    for (int i = 0; i < 4; ++i) {               // B: 64 rows x 64 K
      const int idx = tid + i * 256;
      const int r = idx >> 4;
      const int c = (idx & 15) << 2;
      __builtin_amdgcn_global_load_async_to_lds_b128(
          (AS1 v4i*)(uintptr_t)(Bg + (size_t)r * IN_ + k0 + c),
          (AS3 v4i*)(&lds_b[buf][r * LDP + c]), 0, 0);
    }
  };
  stage_async(0, 0);
  __builtin_amdgcn_s_wait_asynccnt(0);
  __syncthreads();
#else
  float4 ra[8], rb[4];
  auto fetch_regs = [&](int k0) {
    #pragma unroll
    for (int i = 0; i < 8; ++i) {
      const int idx = tid + i * 256;
      const int r = idx >> 4;
      const int c = (idx & 15) << 2;
      ra[i] = *(const float4*)(Ag + (size_t)r * IN_ + k0 + c);
    }
    #pragma unroll
    for (int i = 0; i < 4; ++i) {
      const int idx = tid + i * 256;
      const int r = idx >> 4;
      const int c = (idx & 15) << 2;
      rb[i] = *(const float4*)(Bg + (size_t)r * IN_ + k0 + c);
    }
  };
  auto store_regs = [&](int buf) {
    #pragma unroll
    for (int i = 0; i < 8; ++i) {
      const int idx = tid + i * 256;
      const int r = idx >> 4;
      const int c = (idx & 15) << 2;
      *(float4*)(&lds_a[buf][r * LDP + c]) = ra[i];
    }
    #pragma unroll
    for (int i = 0; i < 4; ++i) {
      const int idx = tid + i * 256;
      const int r = idx >> 4;
      const int c = (idx & 15) << 2;
      *(float4*)(&lds_b[buf][r * LDP + c]) = rb[i];
    }
  };
  fetch_regs(0);
  store_regs(0);
  __syncthreads();
#endif

  for (int k0 = 0; k0 < IN_; k0 += 64) {
    const int cur = (k0 >> 6) & 1;
    const int nxt = cur ^ 1;
    const bool have_next = (k0 + 64) < IN_;

    if (have_next) {
#if USE_ASYNC
      stage_async(k0 + 64, nxt);      // copy next stage while WMMAs run
#else
      fetch_regs(k0 + 64);            // global loads in flight over compute
#endif
    }

    const float* la = lds_a[cur];
    const float* lb = lds_b[cur];
    #pragma unroll
    for (int k = 0; k < 64; k += 4) {
      const v2f a0 = *(const v2f*)(&la[a_row0 + k + koff]);
      const v2f a1 = *(const v2f*)(&la[a_row1 + k + koff]);
      const v2f b0 = *(const v2f*)(&lb[b_row0 + k + koff]);
      const v2f b1 = *(const v2f*)(&lb[b_row1 + k + koff]);
      acc00 = __builtin_amdgcn_wmma_f32_16x16x4_f32(false, a0, false, b0,
                                                    (short)0, acc00, false, false);
      acc01 = __builtin_amdgcn_wmma_f32_16x16x4_f32(false, a0, false, b1,
                                                    (short)0, acc01, false, false);
      acc10 = __builtin_amdgcn_wmma_f32_16x16x4_f32(false, a1, false, b0,
                                                    (short)0, acc10, false, false);
      acc11 = __builtin_amdgcn_wmma_f32_16x16x4_f32(false, a1, false, b1,
                                                    (short)0, acc11, false, false);
    }

    if (have_next) {
#if USE_ASYNC
      __builtin_amdgcn_s_wait_asynccnt(0);
      __syncthreads();
#else
      store_regs(nxt);                // buf[nxt] readers finished last iter
      __syncthreads();
#endif
    }
  }

  // store: C layout -> VGPR v holds row (v + khalf*8), col = mrow within tile
  float* Y = y + ((size_t)b * S_ + m0) * OUT_ + n0;
  #pragma unroll
  for (int v = 0; v < 8; ++v) {
    const int rr = khalf * 8 + v;
    Y[(size_t)(wm * 32 + rr)      * OUT_ + wn * 32 + mrow]      = acc00[v];
    Y[(size_t)(wm * 32 + rr)      * OUT_ + wn * 32 + 16 + mrow] = acc01[v];
    Y[(size_t)(wm * 32 + 16 + rr) * OUT_ + wn * 32 + mrow]      = acc10[v];
    Y[(size_t)(wm * 32 + 16 + rr) * OUT_ + wn * 32 + 16 + mrow] = acc11[v];
  }
}

// ---------------------------------------------------------------------------
extern "C" void kernel_launch(void* const* d_in, const int* in_sizes, int n_in,
                              void* d_out, int out_size, void* d_ws, size_t ws_size,
                              hipStream_t stream) {
  const float* x    = (const float*)d_in[0];
  const float* wg   = (const float*)d_in[1];
  const float* wt   = (const float*)d_in[2];
  const float* resw = (const float*)d_in[3];
  const float* c1i  = (const float*)d_in[4];   // (E,24,24)
  const float* c2i  = (const float*)d_in[5];   // (E,32,32)
  const float* c1o  = (const float*)d_in[6];   // (E,24,24)
  const float* c2o  = (const float*)d_in[7];   // (E,32,32)

  float* y    = (float*)d_out;
  float* loss = y + YSZ;

  float* ws       = (float*)d_ws;
  float* ws_gates = ws;                          // 128
  float* ws_xm    = ws + 128;                    // 12288
  float* bufA     = ws + 16384;                  // E*WSZ
  float* bufB     = bufA + (size_t)E_ * WSZ;     // E*WSZ
  float* ewbuf    = bufB + (size_t)E_ * WSZ;     // B*WSZ

  // gating path
  xm_kernel<<<dim3(B_), dim3(IN_), 0, stream>>>(x, ws_xm);
  gate_kernel<<<dim3(1), dim3(128), 0, stream>>>(ws_xm, wg, ws_gates, loss);

  // curvature transforms: (weight - res) x4 mode products
  mode_strided_kernel<<<dim3(24576 / 32, 1, E_), dim3(256), 0, stream>>>(
      wt, resw, c1o, bufA, 1, 24, 24576);
  mode_strided_kernel<<<dim3(768 / 32, 24, E_), dim3(256), 0, stream>>>(
      bufA, (const float*)nullptr, c2o, bufB, 24, 32, 768);
  mode_strided_kernel<<<dim3(1, 768, E_), dim3(256), 0, stream>>>(
      bufB, (const float*)nullptr, c1i, bufA, 768, 24, 32);
  mode_inner_kernel<<<dim3(18432 / 8, E_), dim3(256), 0, stream>>>(bufA, c2i, bufB);

  // per-batch combined weights
  combine_kernel<<<dim3(WSZ / 256, B_), dim3(256), 0, stream>>>(
      bufB, resw, ws_gates, ewbuf);

  // main batched GEMM with fp32 WMMA
  gemm_kernel<<<dim3(OUT_ / 64, S_ / 128, B_), dim3(256), 0, stream>>>(
      x, ewbuf, y);
}